// GlobalAttentionNet_30210799960809
// MI455X (gfx1250) — compile-verified
//
#include <hip/hip_runtime.h>
#include <hip/hip_bf16.h>
#include <math.h>
#include <stdint.h>

typedef float v2f __attribute__((ext_vector_type(2)));
typedef float v8f __attribute__((ext_vector_type(8)));

#define HID  256
#define NCLS 10
#define NB   64
#define LDA_PAD 4          // dword pad -> lanes map to banks 4i (conflict-free)
#define MAXK 256

// ---------------- utility kernels ----------------

__global__ void fill_f32(float* __restrict__ p, float v, long long n) {
  long long i = (long long)blockIdx.x * blockDim.x + threadIdx.x;
  if (i < n) p[i] = v;
}

__global__ void deg_kernel(const int* __restrict__ dst, float* __restrict__ deg, int E) {
  int e = blockIdx.x * blockDim.x + threadIdx.x;
  if (e < E) atomicAdd(&deg[dst[e]], 1.0f);
}

// scatter-add x[src] rows into agg[dst]; one thread per (edge, 4-float chunk)
__global__ void edge_scatter(const float* __restrict__ x, const int* __restrict__ src,
                             const int* __restrict__ dst, float* __restrict__ agg,
                             long long E, int F, int cshift) {
  long long t = (long long)blockIdx.x * blockDim.x + threadIdx.x;
  long long total = E << cshift;
  if (t >= total) return;
  int e = (int)(t >> cshift);
  int c = (int)(t & ((1 << cshift) - 1));
  int s = src[e], d = dst[e];
  float4 v = *(const float4*)(x + (size_t)s * F + c * 4);
  float* p = agg + (size_t)d * F + c * 4;
  atomicAdd(p + 0, v.x); atomicAdd(p + 1, v.y);
  atomicAdd(p + 2, v.z); atomicAdd(p + 3, v.w);
}

__global__ void scale_by_deg(float* __restrict__ agg, const float* __restrict__ deg,
                             long long total, int fshift) {
  long long t = (long long)blockIdx.x * blockDim.x + threadIdx.x;
  if (t >= total) return;
  int n = (int)(t >> fshift);
  float inv = 1.0f / fmaxf(deg[n], 1.0f);
  agg[t] *= inv;
}

// ---------------- fp32 WMMA GEMM with async-LDS-staged A stripe ----------------
// C[m, j] = relu?( sum_k A1[m,k]*W1[j,k] + (hasA2 ? sum_k A2[m,k]*W2[j,k] : 0) + bias[j] )
// One block = 512 threads = 16 waves; block owns a 16-row x 256-col stripe of C.
// Wave w computes the 16x16 tile at columns [16w, 16w+16). The 16 x K A-tile is
// DMA'd once into LDS via GLOBAL_LOAD_ASYNC_TO_LDS_B128 (ASYNCcnt) and shared by
// all 16 waves (16x global A-traffic reduction).
// A-frag layout (ISA 7.12.2): VGPR = K%2, lane-half = K/2, M = lane&15.
// C layout: VGPR r -> M = r + 8*(lane>>4), N = lane&15.
__global__ __launch_bounds__(512)
void wmma_dual_gemm_lds(const float* __restrict__ A1, const float* __restrict__ W1,
                        const float* __restrict__ A2, const float* __restrict__ W2,
                        const float* __restrict__ bias, float* __restrict__ C,
                        int M, int K, int hasA2, int doRelu) {
  __shared__ float As[16 * (MAXK + LDA_PAD)];
  const int lda = K + LDA_PAD;
  const int tid  = threadIdx.x;
  const int wave = tid >> 5;     // 0..15 == tn tile index
  const int lane = tid & 31;
  const int m0 = blockIdx.x << 4;
  const int n0 = wave << 4;
  const int lr = lane & 15;      // row (A) / col (B,C) within tile
  const int kh = lane >> 4;      // K-half selector
  const int crs = __builtin_ctz((unsigned)K) - 2;   // log2(K/4) chunks per row

  v8f acc = {};

  const int nPass = hasA2 ? 2 : 1;
  for (int pass = 0; pass < nPass; ++pass) {
    const float* A = pass ? A2 : A1;
    const float* W = pass ? W2 : W1;

    if (pass) __syncthreads();   // all waves done reading previous A tile
    // async DMA the 16 x K A tile into LDS (b128 per lane, padded rows)
    int totalChunks = 16 << crs;
    for (int c = tid; c < totalChunks; c += 512) {
      int r = c >> crs;
      int j = (c - (r << crs)) << 2;
      const float* gptr = A + (size_t)(m0 + r) * K + j;
      unsigned lds_off = (unsigned)(uintptr_t)(&As[r * lda + j]);
      asm volatile("global_load_async_to_lds_b128 %0, %1, off"
                   :: "v"(lds_off), "v"(gptr) : "memory");
    }
    asm volatile("s_wait_asynccnt 0x0" ::: "memory");
    __syncthreads();

    const float* as = &As[lr * lda];
    const float* w  = W + (size_t)(n0 + lr) * K;
    for (int k = 0; k < K; k += 4) {
      v2f av, bv;
      av.x = as[k + 2 * kh + 0];
      av.y = as[k + 2 * kh + 1];
      bv.x = w[k + 2 * kh + 0];
      bv.y = w[k + 2 * kh + 1];
      acc = __builtin_amdgcn_wmma_f32_16x16x4_f32(false, av, false, bv,
                                                  (short)0, acc, false, false);
    }
  }

  int col = n0 + lr;
  float bj = bias[col];
#pragma unroll
  for (int r = 0; r < 8; ++r) {
    int row = m0 + r + 8 * kh;
    float v = acc[r] + bj;
    if (doRelu) v = fmaxf(v, 0.0f);
    C[(size_t)row * HID + col] = v;
  }
}

// ---------------- attention pooling ----------------

__global__ void gate_kernel(const float* __restrict__ h, const float* __restrict__ Wg,
                            const float* __restrict__ bg, float* __restrict__ gate, int Nn) {
  int n = blockIdx.x * blockDim.x + threadIdx.x;
  if (n >= Nn) return;
  const float4* hp = (const float4*)(h + (size_t)n * HID);
  const float4* wp = (const float4*)Wg;
  float s = 0.0f;
#pragma unroll 4
  for (int i = 0; i < HID / 4; ++i) {
    float4 a = hp[i], b = wp[i];
    s += a.x * b.x + a.y * b.y + a.z * b.z + a.w * b.w;
  }
  gate[n] = s + bg[0];
}

static __device__ __forceinline__ void atomicMaxF(float* addr, float val) {
  if (val >= 0.0f) atomicMax((int*)addr, __float_as_int(val));
  else             atomicMin((unsigned int*)addr, __float_as_uint(val));
}

__global__ void segmax_kernel(const float* __restrict__ gate, const int* __restrict__ batch,
                              float* __restrict__ m, int Nn) {
  int n = blockIdx.x * blockDim.x + threadIdx.x;
  if (n >= Nn) return;
  atomicMaxF(&m[batch[n]], gate[n]);
}

__global__ void softmax_e_kernel(float* __restrict__ gate, const float* __restrict__ m,
                                 float* __restrict__ s, const int* __restrict__ batch, int Nn) {
  int n = blockIdx.x * blockDim.x + threadIdx.x;
  if (n >= Nn) return;
  int b = batch[n];
  float e = expf(gate[n] - m[b]);
  gate[n] = e;
  atomicAdd(&s[b], e);
}

__global__ void pooled_kernel(const float* __restrict__ h, const float* __restrict__ e,
                              const float* __restrict__ s, const int* __restrict__ batch,
                              float* __restrict__ pooled, int Nn) {
  long long t = (long long)blockIdx.x * blockDim.x + threadIdx.x;
  const int cshift = 6; // HID/4 = 64 chunks
  if (t >= ((long long)Nn << cshift)) return;
  int n = (int)(t >> cshift);
  int c = (int)(t & 63);
  int b = batch[n];
  float a = e[n] / s[b];
  float4 hv = *(const float4*)(h + (size_t)n * HID + c * 4);
  float* p = pooled + (size_t)b * HID + c * 4;
  atomicAdd(p + 0, a * hv.x); atomicAdd(p + 1, a * hv.y);
  atomicAdd(p + 2, a * hv.z); atomicAdd(p + 3, a * hv.w);
}

// ---------------- head ----------------

__global__ void final_linear(const float* __restrict__ in, const float* __restrict__ W,
                             const float* __restrict__ bias, float* __restrict__ out, int Bb) {
  int t = blockIdx.x * blockDim.x + threadIdx.x;
  if (t >= Bb * NCLS) return;
  int b = t / NCLS, c = t % NCLS;
  float s = bias[c];
  for (int k = 0; k < HID; ++k) s += in[b * HID + k] * W[c * HID + k];
  out[t] = s;
}

__global__ void log_softmax_kernel(const float* __restrict__ logits, float* __restrict__ out, int Bb) {
  int b = blockIdx.x * blockDim.x + threadIdx.x;
  if (b >= Bb) return;
  float mx = -1e30f;
  for (int c = 0; c < NCLS; ++c) mx = fmaxf(mx, logits[b * NCLS + c]);
  float s = 0.0f;
  for (int c = 0; c < NCLS; ++c) s += expf(logits[b * NCLS + c] - mx);
  float ls = logf(s);
  for (int c = 0; c < NCLS; ++c) out[b * NCLS + c] = logits[b * NCLS + c] - mx - ls;
}

// ---------------- launcher ----------------

extern "C" void kernel_launch(void* const* d_in, const int* in_sizes, int n_in,
                              void* d_out, int out_size, void* d_ws, size_t ws_size,
                              hipStream_t stream) {
  const float* x     = (const float*)d_in[0];
  const int*   ei    = (const int*)d_in[1];
  const int*   batch = (const int*)d_in[2];
  const float* W1l = (const float*)d_in[3];
  const float* b1l = (const float*)d_in[4];
  const float* W1r = (const float*)d_in[5];
  const float* W2l = (const float*)d_in[6];
  const float* b2l = (const float*)d_in[7];
  const float* W2r = (const float*)d_in[8];
  const float* W3l = (const float*)d_in[9];
  const float* b3l = (const float*)d_in[10];
  const float* W3r = (const float*)d_in[11];
  const float* Wg  = (const float*)d_in[12];
  const float* bg  = (const float*)d_in[13];
  const float* Wl1 = (const float*)d_in[14];
  const float* bl1 = (const float*)d_in[15];
  const float* Wl2 = (const float*)d_in[16];
  const float* bl2 = (const float*)d_in[17];

  const int N = in_sizes[2];
  const long long E = in_sizes[1] / 2;
  const int F_IN = in_sizes[0] / N;       // 128

  const int* src = ei;
  const int* dst = ei + E;

  // workspace partition (256B aligned slices)
  char* ws = (char*)d_ws;
  size_t off = 0;
  auto slice = [&](size_t bytes) -> float* {
    float* p = (float*)(ws + off);
    off += (bytes + 255) & ~(size_t)255;
    return p;
  };
  float* agg    = slice((size_t)N * HID * 4);
  float* hA     = slice((size_t)N * HID * 4);
  float* hB     = slice((size_t)N * HID * 4);
  float* deg    = slice((size_t)N * 4);
  float* gate   = slice((size_t)N * 4);
  float* mB     = slice(NB * 4);
  float* sB     = slice(NB * 4);
  float* pooled = slice(NB * HID * 4);
  float* mlp1   = slice(NB * HID * 4);
  float* logits = slice(NB * NCLS * 4);

  // degree (shared by all layers)
  fill_f32<<<(N + 255) / 256, 256, 0, stream>>>(deg, 0.0f, N);
  deg_kernel<<<(int)((E + 255) / 256), 256, 0, stream>>>(dst, deg, (int)E);

  auto run_layer = [&](const float* xin, int F, int fshift,
                       const float* Wl, const float* bl, const float* Wr, float* hout) {
    long long nAgg = (long long)N * F;
    fill_f32<<<(int)((nAgg + 255) / 256), 256, 0, stream>>>(agg, 0.0f, nAgg);
    long long nE = E << (fshift - 2);
    edge_scatter<<<(int)((nE + 255) / 256), 256, 0, stream>>>(xin, src, dst, agg, E, F, fshift - 2);
    scale_by_deg<<<(int)((nAgg + 255) / 256), 256, 0, stream>>>(agg, deg, nAgg, fshift);
    wmma_dual_gemm_lds<<<N / 16, 512, 0, stream>>>(agg, Wl, xin, Wr, bl, hout, N, F, 1, 1);
  };

  run_layer(x,  F_IN, 7, W1l, b1l, W1r, hA);   // h1 -> hA
  run_layer(hA, HID,  8, W2l, b2l, W2r, hB);   // h2 -> hB
  run_layer(hB, HID,  8, W3l, b3l, W3r, hA);   // h3 -> hA (h1 dead)

  // attention pooling
  gate_kernel<<<(N + 255) / 256, 256, 0, stream>>>(hA, Wg, bg, gate, N);
  fill_f32<<<1, 64, 0, stream>>>(mB, -INFINITY, NB);
  fill_f32<<<1, 64, 0, stream>>>(sB, 0.0f, NB);
  fill_f32<<<(NB * HID + 255) / 256, 256, 0, stream>>>(pooled, 0.0f, NB * HID);
  segmax_kernel<<<(N + 255) / 256, 256, 0, stream>>>(gate, batch, mB, N);
  softmax_e_kernel<<<(N + 255) / 256, 256, 0, stream>>>(gate, mB, sB, batch, N);
  long long nPool = (long long)N * (HID / 4);
  pooled_kernel<<<(int)((nPool + 255) / 256), 256, 0, stream>>>(hA, gate, sB, batch, pooled, N);

  // MLP head: WMMA for [64x256]x[256x256], scalar for tiny final layer
  wmma_dual_gemm_lds<<<NB / 16, 512, 0, stream>>>(pooled, Wl1, nullptr, nullptr, bl1, mlp1,
                                                  NB, HID, 0, 1);
  final_linear<<<(NB * NCLS + 255) / 256, 256, 0, stream>>>(mlp1, Wl2, bl2, logits, NB);
  log_softmax_kernel<<<1, 64, 0, stream>>>(logits, (float*)d_out, NB);
}